// DiffDistMatrix_86887188398725
// MI455X (gfx1250) — compile-verified
//
#include <hip/hip_runtime.h>
#include <cmath>

// S=64 cells, I=128 points, 27 periodic images.
//   out0: dist[S,I,I], out1: cart1[S,I,3]
// dist^2[i,j,m] = |a_i + c_m|^2 + |b_j|^2 - 2 a_i.b_j - 2 b_j.c_m
// Cross term a.b (128x128, K=3) via V_WMMA_F32_16X16X4_F32; the 27-way min
// runs on transposed LDS tables with b128 broadcast loads.

typedef __attribute__((ext_vector_type(2))) float v2f;
typedef __attribute__((ext_vector_type(4))) float v4f;
typedef __attribute__((ext_vector_type(8))) float v8f;

#define S_DIM 64
#define I_DIM 128
#define NIMG  27
#define QSPLIT 4                 // blockIdx.y groups of column tiles
#define TJ_PER_Q (8 / QSPLIT)    // column tiles per block per wave

__global__ __launch_bounds__(256)
void diffdist_wmma_kernel(const float* __restrict__ fc1,
                          const float* __restrict__ fc2,
                          const float* __restrict__ mat,
                          float* __restrict__ out_dist,
                          float* __restrict__ out_cart)
{
    const int s   = blockIdx.x;
    const int q   = blockIdx.y;
    const int tid = threadIdx.x;

    __shared__ __align__(16) float a3[I_DIM][3];     // cart2 (rows i)
    __shared__ __align__(16) float b3[I_DIM][3];     // cart1 (cols j)
    __shared__ __align__(16) float cim[NIMG][3];     // image offsets (cartesian)
    __shared__ __align__(16) float B2[I_DIM];        // |b_j|^2
    __shared__ __align__(16) float Tt[NIMG][I_DIM];  // |a_i + c_m|^2   (transposed)
    __shared__ __align__(16) float Ut[NIMG][I_DIM];  // -2 b_j . c_m    (transposed)

    const float* M = mat + s * 9;
    const float m00 = M[0], m01 = M[1], m02 = M[2];
    const float m10 = M[3], m11 = M[4], m12 = M[5];
    const float m20 = M[6], m21 = M[7], m22 = M[8];

    // ---- Phase 1: fractional -> cartesian ----------------------------------
    {
        const int i = tid & 127;
        const float* f = (tid < 128 ? fc1 : fc2) + ((size_t)s * I_DIM + i) * 3;
        float f0 = f[0]; f0 -= floorf(f0);
        float f1 = f[1]; f1 -= floorf(f1);
        float f2 = f[2]; f2 -= floorf(f2);
        const float cx = f0 * m00 + f1 * m10 + f2 * m20;
        const float cy = f0 * m01 + f1 * m11 + f2 * m21;
        const float cz = f0 * m02 + f1 * m12 + f2 * m22;
        if (tid < 128) {
            b3[i][0] = cx; b3[i][1] = cy; b3[i][2] = cz;
            if (q == 0) {
                float* o = out_cart + ((size_t)s * I_DIM + i) * 3;
                o[0] = cx; o[1] = cy; o[2] = cz;
            }
        } else {
            a3[i][0] = cx; a3[i][1] = cy; a3[i][2] = cz;
        }
    }
    if (tid < NIMG) {
        const float ia = (float)(tid / 9) - 1.0f;
        const float ib = (float)((tid / 3) % 3) - 1.0f;
        const float ic = (float)(tid % 3) - 1.0f;
        cim[tid][0] = ia * m00 + ib * m10 + ic * m20;
        cim[tid][1] = ia * m01 + ib * m11 + ic * m21;
        cim[tid][2] = ia * m02 + ib * m12 + ic * m22;
    }
    __syncthreads();

    // ---- Phase 2: per-point image tables (transposed layout) ---------------
    {
        const int i = tid & 127;
        if (tid < 128) {
            const float bx = b3[i][0], by = b3[i][1], bz = b3[i][2];
            B2[i] = bx * bx + by * by + bz * bz;
            for (int m = 0; m < NIMG; ++m)
                Ut[m][i] = -2.0f * (bx * cim[m][0] + by * cim[m][1] + bz * cim[m][2]);
        } else {
            const float ax = a3[i][0], ay = a3[i][1], az = a3[i][2];
            for (int m = 0; m < NIMG; ++m) {
                const float dx = ax + cim[m][0];
                const float dy = ay + cim[m][1];
                const float dz = az + cim[m][2];
                Tt[m][i] = dx * dx + dy * dy + dz * dz;
            }
        }
    }
    __syncthreads();

    // ---- Phase 3: WMMA cross-term + 27-way min -----------------------------
    const int wv   = tid >> 5;
    const int lane = tid & 31;
    const int half = lane >> 4;
    const int l15  = lane & 15;
    const int ti   = wv;

    // A operand (16x4, K padded 3->4): lanes 0-15 carry (K0,K1), lanes 16-31 (K2,0)
    const int arow = ti * 16 + l15;
    v2f A;
    A.x = half ? a3[arow][2] : a3[arow][0];
    A.y = half ? 0.0f        : a3[arow][1];

    const int ibase = ti * 16 + half * 8;   // C/D rows: M = r + 8*half (32B aligned)

    for (int t = 0; t < TJ_PER_Q; ++t) {
        const int tj = q * TJ_PER_Q + t;
        const int j  = tj * 16 + l15;
        v2f B;
        B.x = half ? b3[j][2] : b3[j][0];
        B.y = half ? 0.0f     : b3[j][1];

        v8f c = {};
        c = __builtin_amdgcn_wmma_f32_16x16x4_f32(
                false, A, false, B, (short)0, c, false, false);

        const float b2j = B2[j];

        float dmin[8];
        #pragma unroll
        for (int r = 0; r < 8; ++r) dmin[r] = 3.4e38f;

        for (int m = 0; m < NIMG; ++m) {
            const float u = Ut[m][j];                       // conflict-free
            const v4f t0 = *(const v4f*)&Tt[m][ibase];      // ds_load_b128 (broadcast)
            const v4f t1 = *(const v4f*)&Tt[m][ibase + 4];
            dmin[0] = fminf(dmin[0], t0.x + u);
            dmin[1] = fminf(dmin[1], t0.y + u);
            dmin[2] = fminf(dmin[2], t0.z + u);
            dmin[3] = fminf(dmin[3], t0.w + u);
            dmin[4] = fminf(dmin[4], t1.x + u);
            dmin[5] = fminf(dmin[5], t1.y + u);
            dmin[6] = fminf(dmin[6], t1.z + u);
            dmin[7] = fminf(dmin[7], t1.w + u);
        }

        #pragma unroll
        for (int r = 0; r < 8; ++r) {
            float d2 = dmin[r] + b2j - 2.0f * c[r] + 1e-10f;
            out_dist[((size_t)s * I_DIM + (ibase + r)) * I_DIM + j] =
                __builtin_amdgcn_sqrtf(fmaxf(d2, 0.0f));    // bare v_sqrt_f32
        }
    }
}

extern "C" void kernel_launch(void* const* d_in, const int* in_sizes, int n_in,
                              void* d_out, int out_size, void* d_ws, size_t ws_size,
                              hipStream_t stream) {
    const float* fc1 = (const float*)d_in[0];   // fcoords1 [64,128,3]
    const float* fc2 = (const float*)d_in[1];   // fcoords2 [64,128,3]
    const float* mat = (const float*)d_in[2];   // matrix   [64,3,3]
    float* out_dist = (float*)d_out;                               // [64,128,128]
    float* out_cart = out_dist + (size_t)S_DIM * I_DIM * I_DIM;    // [64,128,3]

    diffdist_wmma_kernel<<<dim3(S_DIM, QSPLIT), dim3(256), 0, stream>>>(
        fc1, fc2, mat, out_dist, out_cart);
    (void)in_sizes; (void)n_in; (void)out_size; (void)d_ws; (void)ws_size;
}